// Embedding_61864708932005
// MI455X (gfx1250) — compile-verified
//
#include <hip/hip_runtime.h>

// ---------------------------------------------------------------------------
// Fused masked marker-embedding:
//   out[s,:] = (t[s] < 0) ? 0 : 0.5*W_m[:,marker[s]] + 0.5*(t[s]*W_t + b_t)
// Memory-bound: 256 MB output stream; gather served from L2-resident W_m^T.
// ---------------------------------------------------------------------------

typedef float v4f __attribute__((ext_vector_type(4)));

#define D_MODEL 512
#define N_MARK  1000
#define CHUNK   64     // rows staged per block
#define BLOCK   256    // 8 wave32; 128 threads cover one 512-float row as float4

typedef __attribute__((address_space(1))) int g1_int;   // global (AS1)
typedef __attribute__((address_space(3))) int l3_int;   // LDS (AS3)

// --- gfx1250 async global->LDS staging (guarded; falls back to plain load) --
__device__ __forceinline__ void stage_b32(const void* g, void* lds) {
#if __has_builtin(__builtin_amdgcn_global_load_async_to_lds_b32)
  __builtin_amdgcn_global_load_async_to_lds_b32(
      (g1_int*)g, (l3_int*)lds, /*offset=*/0, /*cpol=*/0);
#else
  *(int*)lds = *(const int*)g;
#endif
}

__device__ __forceinline__ void wait_async_stage() {
#if __has_builtin(__builtin_amdgcn_global_load_async_to_lds_b32)
 #if __has_builtin(__builtin_amdgcn_s_wait_asynccnt)
  __builtin_amdgcn_s_wait_asynccnt(0);
 #else
  asm volatile("s_wait_asynccnt 0" ::: "memory");
 #endif
#endif
}

// --- main fused streaming kernel (first in file -> shows in disasm snippet) -
__global__ __launch_bounds__(BLOCK) void fused_embed(
    const float* __restrict__ t, const int* __restrict__ marker,
    const float* __restrict__ Wg,   // W_mT (transposed) or W_m (fallback)
    const float* __restrict__ Wt, const float* __restrict__ bt,
    float* __restrict__ out, int S, int transposed) {
  __shared__ float s_t[CHUNK];
  __shared__ int   s_m[CHUNK];

  const int  tid  = threadIdx.x;
  const int  d4   = tid & 127;        // float4 column within the row
  const int  sub  = tid >> 7;         // 0/1: which of the 2 rows per iter
  const long base = (long)blockIdx.x * CHUNK;

  // stage this block's t / marker slice into LDS (async on gfx1250)
  if (tid < CHUNK && base + tid < S) {
    stage_b32(t + base + tid, &s_t[tid]);
    stage_b32(marker + base + tid, &s_m[tid]);
  }
  wait_async_stage();
  __syncthreads();

  const v4f wt = ((const v4f*)Wt)[d4];   // per-thread constants, held in VGPRs
  const v4f bv = ((const v4f*)bt)[d4];

#pragma unroll 8
  for (int i = 0; i < CHUNK; i += 2) {
    const int  lr  = i + sub;
    const long row = base + lr;
    if (row < S) {
      const float tv = s_t[lr];         // LDS broadcast (uniform per wave)
      const int   mk = s_m[lr];

      v4f r = {0.0f, 0.0f, 0.0f, 0.0f};
      if (tv >= 0.0f) {
        v4f g;
        if (transposed) {
          // contiguous 2KB row of W_mT: coalesced, L2/WGP$ resident
          g = ((const v4f*)Wg)[(long)mk * (D_MODEL / 4) + d4];
        } else {
          const float* col = Wg + mk;   // strided fallback (no workspace)
          const int d = d4 * 4;
          g.x = col[(long)(d + 0) * N_MARK];
          g.y = col[(long)(d + 1) * N_MARK];
          g.z = col[(long)(d + 2) * N_MARK];
          g.w = col[(long)(d + 3) * N_MARK];
        }
        r = 0.5f * g + 0.5f * (tv * wt + bv);
      }
      // write-once 256MB stream: non-temporal b128 store, keep L2 for W_mT
      __builtin_nontemporal_store(r, (v4f*)out + row * (D_MODEL / 4) + d4);
    }
  }
}

// --- one-time LDS-tiled transpose: W_m[512][1000] -> W_mT[1000][512] --------
__global__ __launch_bounds__(256) void transpose_wm(
    const float* __restrict__ Wm, float* __restrict__ WmT) {
  __shared__ float tile[32][33];
  const int bm = blockIdx.x * 32;     // marker tile
  const int bd = blockIdx.y * 32;     // d_model tile
  const int tx = threadIdx.x;         // 0..31
  const int ty = threadIdx.y;         // 0..7
#pragma unroll
  for (int i = 0; i < 32; i += 8) {
    const int d = bd + ty + i;
    const int m = bm + tx;
    if (d < D_MODEL && m < N_MARK)
      tile[ty + i][tx] = Wm[(long)d * N_MARK + m];   // coalesced read
  }
  __syncthreads();
#pragma unroll
  for (int i = 0; i < 32; i += 8) {
    const int m = bm + ty + i;
    const int d = bd + tx;
    if (m < N_MARK && d < D_MODEL)
      WmT[(long)m * D_MODEL + d] = tile[tx][ty + i]; // coalesced write
  }
}

extern "C" void kernel_launch(void* const* d_in, const int* in_sizes, int n_in,
                              void* d_out, int out_size, void* d_ws,
                              size_t ws_size, hipStream_t stream) {
  const float* t      = (const float*)d_in[0];
  const int*   marker = (const int*)d_in[1];
  const float* Wm     = (const float*)d_in[2];
  const float* Wt     = (const float*)d_in[3];
  const float* bt     = (const float*)d_in[4];
  float*       out    = (float*)d_out;
  const int    S      = in_sizes[0];

  const size_t need = (size_t)N_MARK * D_MODEL * sizeof(float);
  const int transposed = (d_ws != nullptr && ws_size >= need) ? 1 : 0;

  const float* Wg = Wm;
  if (transposed) {
    float* WmT = (float*)d_ws;
    dim3 tgrid((N_MARK + 31) / 32, D_MODEL / 32);
    transpose_wm<<<tgrid, dim3(32, 8), 0, stream>>>(Wm, WmT);
    Wg = WmT;
  }

  const int nblocks = (S + CHUNK - 1) / CHUNK;
  fused_embed<<<nblocks, BLOCK, 0, stream>>>(t, marker, Wg, Wt, bt, out, S,
                                             transposed);
}